// VideoSaliencyLoss_70171175682840
// MI455X (gfx1250) — compile-verified
//
#include <hip/hip_runtime.h>
#include <stdint.h>
#include <math.h>

typedef float v2f __attribute__((ext_vector_type(2)));
typedef float v8f __attribute__((ext_vector_type(8)));

#define FRAME 921600            // 720*1280
#define FRAME4 230400           // FRAME/4
#define NFRAMES 32
#define NALL_F 29491200.0f
#define BPF 180                 // blocks per frame in streaming kernels
#define ITERS 5                 // 180 * 5 * 1024 = 921600

// ---- workspace layout (bytes) ----
#define OFF_HIST   0ULL         // u32[65536]
#define OFF_BASE   262144ULL    // u32[65537]
#define OFF_CNT    532480ULL    // u32[65536]
#define OFF_SCAL   794624ULL    // +0 pmin_u, +4 gmin_u, +8 Rmax, +16 Ssum(u64), +24 sc[4] floats
#define OFF_PARTA  795648ULL    // float[32][180][5]
#define OFF_SIMP   912384ULL    // float[32][180]
#define OFF_FS     935936ULL    // float[32][5]
#define OFF_ENT    942080ULL    // u64[921600]

__device__ __forceinline__ unsigned mapf(float f) {
  unsigned u = __float_as_uint(f);
  return u ^ ((unsigned)(((int)u) >> 31) | 0x80000000u);
}
__device__ __forceinline__ float unmapf(unsigned u) {
  unsigned b = (u & 0x80000000u) ? (u ^ 0x80000000u) : ~u;
  return __uint_as_float(b);
}
__device__ __forceinline__ unsigned uminu(unsigned a, unsigned b) { return a < b ? a : b; }

// f32 WMMA row-sum accumulate: D = A x ones + C  (exact f32 FMA, RNE)
__device__ __forceinline__ v8f wacc(v2f a, v8f c) {
  v2f ones = {1.0f, 1.0f};
  return __builtin_amdgcn_wmma_f32_16x16x4_f32(false, a, false, ones, (short)0, c, false, false);
}
__device__ __forceinline__ float sum8(v8f c) {
  return ((c[0] + c[1]) + (c[2] + c[3])) + ((c[4] + c[5]) + (c[6] + c[7]));
}

// ---------------- init ----------------
__global__ void k_init(unsigned char* ws) {
  unsigned i = blockIdx.x * blockDim.x + threadIdx.x;
  unsigned* hist = (unsigned*)(ws + OFF_HIST);
  unsigned* cnt  = (unsigned*)(ws + OFF_CNT);
  if (i < 65536) hist[i] = 0u;
  else if (i < 131072) cnt[i - 65536] = 0u;
  if (i == 0) {
    *(unsigned*)(ws + OFF_SCAL + 0) = 0xFFFFFFFFu;
    *(unsigned*)(ws + OFF_SCAL + 4) = 0xFFFFFFFFu;
    *(unsigned*)(ws + OFF_SCAL + 8) = 0u;
    *(unsigned long long*)(ws + OFF_SCAL + 16) = 0ull;
  }
}

// ---------------- pass A: per-frame moments + masked sums + global mins ----------------
__global__ void __launch_bounds__(256) k_reduceA(const float* __restrict__ pred,
                                                 const float* __restrict__ gt,
                                                 const int* __restrict__ fix,
                                                 unsigned char* ws) {
  const int f = blockIdx.y, b = blockIdx.x, tid = threadIdx.x;
  const float4* p4 = (const float4*)pred;
  const float4* g4 = (const float4*)gt;
  const int4*   x4 = (const int4*)fix;
  size_t base4 = (size_t)f * FRAME4 + (size_t)b * (256 * ITERS);
  v8f cp = {}, cpp = {}, cmp = {}, cmc = {}, cg = {};
  float pminl = INFINITY, gminl = INFINITY;
  for (int it = 0; it < ITERS; ++it) {
    size_t i = base4 + (size_t)it * 256 + tid;
    float4 p = p4[i];
    float4 g = g4[i];
    int4   m = x4[i];
    v2f ap0 = {p.x, p.y};
    v2f ap1 = {p.z, p.w};
    v2f aq0 = {p.x * p.x, p.y * p.y};
    v2f aq1 = {p.z * p.z, p.w * p.w};
    v2f am0 = {m.x != 0 ? p.x : 0.f, m.y != 0 ? p.y : 0.f};
    v2f am1 = {m.z != 0 ? p.z : 0.f, m.w != 0 ? p.w : 0.f};
    v2f ac0 = {m.x != 0 ? 1.f : 0.f, m.y != 0 ? 1.f : 0.f};
    v2f ac1 = {m.z != 0 ? 1.f : 0.f, m.w != 0 ? 1.f : 0.f};
    v2f ag0 = {g.x, g.y};
    v2f ag1 = {g.z, g.w};
    cp  = wacc(ap0, cp);   cp  = wacc(ap1, cp);
    cpp = wacc(aq0, cpp);  cpp = wacc(aq1, cpp);
    cmp = wacc(am0, cmp);  cmp = wacc(am1, cmp);
    cmc = wacc(ac0, cmc);  cmc = wacc(ac1, cmc);
    cg  = wacc(ag0, cg);   cg  = wacc(ag1, cg);
    pminl = fminf(pminl, fminf(fminf(p.x, p.y), fminf(p.z, p.w)));
    gminl = fminf(gminl, fminf(fminf(g.x, g.y), fminf(g.z, g.w)));
  }
  __shared__ float red[256];
  __shared__ unsigned ured[256];
  int lane = tid & 31;
  bool keep = (lane == 0) || (lane == 16);
  float contrib[5];
  contrib[0] = keep ? sum8(cp)  : 0.f;
  contrib[1] = keep ? sum8(cpp) : 0.f;
  contrib[2] = keep ? sum8(cmp) : 0.f;
  contrib[3] = keep ? sum8(cmc) : 0.f;
  contrib[4] = keep ? sum8(cg)  : 0.f;
  float* partA = (float*)(ws + OFF_PARTA);
  for (int q = 0; q < 5; ++q) {
    red[tid] = contrib[q];
    __syncthreads();
    for (int s = 128; s; s >>= 1) { if (tid < s) red[tid] += red[tid + s]; __syncthreads(); }
    if (tid == 0) partA[((size_t)f * BPF + b) * 5 + q] = red[0];
    __syncthreads();
  }
  ured[tid] = mapf(pminl); __syncthreads();
  for (int s = 128; s; s >>= 1) { if (tid < s) ured[tid] = uminu(ured[tid], ured[tid + s]); __syncthreads(); }
  if (tid == 0) atomicMin((unsigned*)(ws + OFF_SCAL + 0), ured[0]);
  __syncthreads();
  ured[tid] = mapf(gminl); __syncthreads();
  for (int s = 128; s; s >>= 1) { if (tid < s) ured[tid] = uminu(ured[tid], ured[tid + s]); __syncthreads(); }
  if (tid == 0) atomicMin((unsigned*)(ws + OFF_SCAL + 4), ured[0]);
}

// ---------------- combine per-frame partials (fixed-order, deterministic) ----------------
__global__ void k_combineA(unsigned char* ws) {
  int f = blockIdx.x, tid = threadIdx.x;
  const float* partA = (const float*)(ws + OFF_PARTA);
  float* FS = (float*)(ws + OFF_FS);
  __shared__ float red[256];
  for (int q = 0; q < 5; ++q) {
    red[tid] = (tid < BPF) ? partA[((size_t)f * BPF + tid) * 5 + q] : 0.f;
    __syncthreads();
    for (int s = 128; s; s >>= 1) { if (tid < s) red[tid] += red[tid + s]; __syncthreads(); }
    if (tid == 0) FS[f * 5 + q] = red[0];
    __syncthreads();
  }
}

// ---------------- derive SIM scalars ----------------
__global__ void k_scalars(unsigned char* ws) {
  const float* FS = (const float*)(ws + OFF_FS);
  float* sc = (float*)(ws + OFF_SCAL + 24);
  float sumP = 0.f, sumG = 0.f;
  for (int f = 0; f < NFRAMES; ++f) { sumP += FS[f * 5 + 0]; sumG += FS[f * 5 + 4]; }
  float pminf = unmapf(*(const unsigned*)(ws + OFF_SCAL + 0));
  float gminf = unmapf(*(const unsigned*)(ws + OFF_SCAL + 4));
  float Dp = sumP - NALL_F * pminf;   // (max-min) cancels in normalized values
  float Dg = sumG - NALL_F * gminf;
  sc[0] = pminf; sc[1] = gminf; sc[2] = 1.f / Dp; sc[3] = 1.f / Dg;
}

// ---------------- pass B: SIM histogram-overlap ----------------
__global__ void __launch_bounds__(256) k_sim(const float* __restrict__ pred,
                                             const float* __restrict__ gt,
                                             unsigned char* ws) {
  const int f = blockIdx.y, b = blockIdx.x, tid = threadIdx.x;
  const float* sc = (const float*)(ws + OFF_SCAL + 24);
  float pminf = sc[0], gminf = sc[1], invDp = sc[2], invDg = sc[3];
  const float4* p4 = (const float4*)pred;
  const float4* g4 = (const float4*)gt;
  size_t base4 = (size_t)f * FRAME4 + (size_t)b * (256 * ITERS);
  v8f cs = {};
  for (int it = 0; it < ITERS; ++it) {
    size_t i = base4 + (size_t)it * 256 + tid;
    float4 p = p4[i];
    float4 g = g4[i];
    v2f a0, a1;
    a0.x = (g.x > gminf) ? fminf((g.x - gminf) * invDg, (p.x - pminf) * invDp) : 0.f;
    a0.y = (g.y > gminf) ? fminf((g.y - gminf) * invDg, (p.y - pminf) * invDp) : 0.f;
    a1.x = (g.z > gminf) ? fminf((g.z - gminf) * invDg, (p.z - pminf) * invDp) : 0.f;
    a1.y = (g.w > gminf) ? fminf((g.w - gminf) * invDg, (p.w - pminf) * invDp) : 0.f;
    cs = wacc(a0, cs);
    cs = wacc(a1, cs);
  }
  __shared__ float red[256];
  int lane = tid & 31;
  bool keep = (lane == 0) || (lane == 16);
  red[tid] = keep ? sum8(cs) : 0.f;
  __syncthreads();
  for (int s = 128; s; s >>= 1) { if (tid < s) red[tid] += red[tid + s]; __syncthreads(); }
  if (tid == 0) ((float*)(ws + OFF_SIMP))[(size_t)f * BPF + b] = red[0];
}

// ---------------- AUC: histogram of top-16 monotone key bits (frame 0) ----------------
__global__ void k_hist(const float* __restrict__ pred, unsigned char* ws) {
  unsigned* hist = (unsigned*)(ws + OFF_HIST);
  unsigned i = blockIdx.x * 1024u + threadIdx.x;    // 900*1024 = 921600
  unsigned u = mapf(pred[i]);
  atomicAdd(&hist[u >> 16], 1u);
}

// ---------------- exclusive scan of 65536 bins (single block) ----------------
__global__ void k_scan(unsigned char* ws) {
  const unsigned* hist = (const unsigned*)(ws + OFF_HIST);
  unsigned* base = (unsigned*)(ws + OFF_BASE);
  __shared__ unsigned s[1024];
  __shared__ unsigned carry;
  int tid = threadIdx.x;
  if (tid == 0) carry = 0u;
  __syncthreads();
  for (int c = 0; c < 64; ++c) {
    unsigned h = hist[c * 1024 + tid];
    s[tid] = h;
    __syncthreads();
    for (int off = 1; off < 1024; off <<= 1) {
      unsigned t = (tid >= off) ? s[tid - off] : 0u;
      __syncthreads();
      if (tid >= off) s[tid] += t;
      __syncthreads();
    }
    unsigned c0 = carry;
    unsigned last = s[1023];
    base[c * 1024 + tid] = c0 + s[tid] - h;   // exclusive
    __syncthreads();
    if (tid == 0) carry = c0 + last;
    __syncthreads();
  }
  if (tid == 0) base[65536] = carry;
}

// ---------------- scatter packed (low16, inv_idx, fixflag) into bucket-sorted order ----------------
__global__ void k_scatter(const float* __restrict__ pred, const int* __restrict__ fix,
                          unsigned char* ws) {
  const unsigned* base = (const unsigned*)(ws + OFF_BASE);
  unsigned* cnt = (unsigned*)(ws + OFF_CNT);
  unsigned long long* ent = (unsigned long long*)(ws + OFF_ENT);
  unsigned i = blockIdx.x * 1024u + threadIdx.x;
  unsigned u = mapf(pred[i]);
  unsigned b = u >> 16;
  unsigned pos = base[b] + atomicAdd(&cnt[b], 1u);
  unsigned long long low  = (unsigned long long)(u & 0xFFFFu);
  unsigned long long inv  = (unsigned long long)(0xFFFFFu - i);   // inverted index: stable tie-break
  unsigned long long fb   = (fix[i] != 0) ? 1ull : 0ull;
  ent[pos] = (low << 21) | (inv << 1) | fb;
}

// ---------------- per-bucket rank sum/max over fixation pixels ----------------
#define CAP 4096
__global__ void __launch_bounds__(128) k_rank(unsigned char* ws) {
  const unsigned* base = (const unsigned*)(ws + OFF_BASE);
  const unsigned long long* ent = (const unsigned long long*)(ws + OFF_ENT);
  unsigned b = blockIdx.x, tid = threadIdx.x;
  unsigned start = base[b], end = base[b + 1], n = end - start;
  if (n == 0) return;                         // uniform per block
  __shared__ unsigned long long sh[CAP];
  __shared__ unsigned long long rs[128];
  __shared__ unsigned rr[128];
  bool lds = (n <= CAP);
  if (lds) for (unsigned i = tid; i < n; i += 128) sh[i] = ent[start + i];
  __syncthreads();
  unsigned long long S = 0ull;
  unsigned R = 0u;
  unsigned long long baseRank = (unsigned long long)FRAME - (unsigned long long)end;
  for (unsigned i = tid; i < n; i += 128) {
    unsigned long long e = lds ? sh[i] : ent[start + i];
    if (e & 1ull) {
      unsigned long long me = e >> 1;
      unsigned cgt = 0;
      for (unsigned j = 0; j < n; ++j) {
        unsigned long long o = (lds ? sh[j] : ent[start + j]) >> 1;
        cgt += (o > me) ? 1u : 0u;
      }
      unsigned long long rank = baseRank + (unsigned long long)cgt + 1ull;  // descending, stable
      S += rank;
      unsigned r32 = (unsigned)rank;
      if (r32 > R) R = r32;
    }
  }
  rs[tid] = S; rr[tid] = R;
  __syncthreads();
  for (int s = 64; s; s >>= 1) {
    if ((int)tid < s) { rs[tid] += rs[tid + s]; rr[tid] = rr[tid] > rr[tid + s] ? rr[tid] : rr[tid + s]; }
    __syncthreads();
  }
  if (tid == 0) {
    if (rs[0]) atomicAdd((unsigned long long*)(ws + OFF_SCAL + 16), rs[0]);
    atomicMax((unsigned*)(ws + OFF_SCAL + 8), rr[0]);
  }
}

// ---------------- final: NSS + SIM mean + closed-form AUC ----------------
__global__ void k_final(unsigned char* ws, float* out) {
  __shared__ float red[256];
  __shared__ float simF[NFRAMES];
  int tid = threadIdx.x;
  const float* simPart = (const float*)(ws + OFF_SIMP);
  for (int f = 0; f < NFRAMES; ++f) {
    red[tid] = (tid < BPF) ? simPart[(size_t)f * BPF + tid] : 0.f;
    __syncthreads();
    for (int s = 128; s; s >>= 1) { if (tid < s) red[tid] += red[tid + s]; __syncthreads(); }
    if (tid == 0) simF[f] = red[0];
    __syncthreads();
  }
  if (tid == 0) {
    const float* FS = (const float*)(ws + OFF_FS);
    float simAcc = 0.f;
    for (int f = 0; f < NFRAMES; ++f) simAcc += simF[f];
    float sim = simAcc / (float)NFRAMES;
    float nssAcc = 0.f;
    const float invN = 1.f / (float)FRAME;
    for (int f = 0; f < NFRAMES; ++f) {
      float sp  = FS[f * 5 + 0];
      float spp = FS[f * 5 + 1];
      float sfp = FS[f * 5 + 2];
      float cnt = FS[f * 5 + 3];
      float mean = sp * invN;
      float var  = spp * invN - mean * mean;
      float stdv = sqrtf(var);
      nssAcc += (sfp - cnt * mean) / (stdv * cnt);
    }
    float nss = nssAcc / (float)NFRAMES;
    double Nf = (double)FS[3];               // frame-0 fixation count (exact in f32)
    double Np = (double)FRAME;
    unsigned long long S = *(const unsigned long long*)(ws + OFF_SCAL + 16);
    unsigned R = *(const unsigned*)(ws + OFF_SCAL + 8);
    double denom = Np - Nf;
    double fpN   = ((double)R - (Nf - 1.0)) / denom;
    double sumfp = ((double)S - Nf * (Nf - 1.0) * 0.5) / denom;
    double auc   = 1.0 - (sumfp - 0.5 * fpN) / Nf;
    float aucf = (float)auc;
    out[0] = aucf + sim + nss;
    out[1] = aucf;
    out[2] = sim;
    out[3] = nss;
  }
}

extern "C" void kernel_launch(void* const* d_in, const int* in_sizes, int n_in,
                              void* d_out, int out_size, void* d_ws, size_t ws_size,
                              hipStream_t stream) {
  (void)in_sizes; (void)n_in; (void)out_size; (void)ws_size;
  const float* pred = (const float*)d_in[0];
  const float* gt   = (const float*)d_in[1];
  const int*   fix  = (const int*)d_in[2];
  unsigned char* ws = (unsigned char*)d_ws;
  float* out = (float*)d_out;

  k_init<<<512, 256, 0, stream>>>(ws);
  dim3 gA(BPF, NFRAMES);
  k_reduceA<<<gA, 256, 0, stream>>>(pred, gt, fix, ws);
  k_combineA<<<NFRAMES, 256, 0, stream>>>(ws);
  k_scalars<<<1, 1, 0, stream>>>(ws);
  k_sim<<<gA, 256, 0, stream>>>(pred, gt, ws);
  k_hist<<<900, 1024, 0, stream>>>(pred, ws);      // frame 0
  k_scan<<<1, 1024, 0, stream>>>(ws);
  k_scatter<<<900, 1024, 0, stream>>>(pred, fix, ws);
  k_rank<<<65536, 128, 0, stream>>>(ws);
  k_final<<<1, 256, 0, stream>>>(ws, out);
}